// Simulator_58377195487948
// MI455X (gfx1250) — compile-verified
//
#include <hip/hip_runtime.h>
#include <math.h>

typedef __attribute__((ext_vector_type(2))) float v2f;
typedef __attribute__((ext_vector_type(8))) float v8f;

#define LMAX   180
#define LDIM   181      // L
#define TL     361      // 2L-1
#define NF     50
#define NT     240
#define NCOL   100      // 2*NF (re/im interleaved freq columns)
#define MT     23       // ceil(361/16) row tiles
#define SPITCH 368      // padded m-pitch for S accumulators
#define XTP    388      // transposed-X pitch (words); 388 % 64 == 4 -> conflict-free b64

// ---------------- kernel 1: zero the accumulators in workspace ----------------
__global__ void sim_zero_kernel(float* __restrict__ ws, int n) {
    int i = blockIdx.x * blockDim.x + threadIdx.x;
    if (i < n) ws[i] = 0.0f;
}

// ---- epilogue: pa z-phase, conj(beam_eq)*sky, atomic accumulation over l ----
__device__ __forceinline__ void sim_epilogue(
    v8f acc, int m0, int laneHi, int gcol, int l, float alpha,
    const float* __restrict__ sky_re, const float* __restrict__ sky_im,
    float* __restrict__ Sre, float* __restrict__ Sim_, float* __restrict__ nrm)
{
    const int isIm = gcol & 1;
    #pragma unroll
    for (int j = 0; j < 8; ++j) {
        int drow = m0 + j + 8 * laneHi;                 // D row per ISA C/D layout
        float y  = acc[j];
        float yp = __shfl_xor(y, 1, 32);                // re/im partner column
        float Yre = isIm ? yp : y;
        float Yim = isIm ? y  : yp;
        float sA, cA;
        __sincosf((float)(drow - LMAX) * alpha, &sA, &cA);
        // pa = cA - i sA : be_re = cA*Yre + sA*Yim ; be_im = cA*Yim - sA*Yre
        float be  = isIm ? (cA * Yim - sA * Yre) : (cA * Yre + sA * Yim);
        float bep = __shfl_xor(be, 1, 32);
        float be_re = isIm ? bep : be;
        float be_im = isIm ? be  : bep;
        if (drow < TL && gcol < NCOL) {
            int f = gcol >> 1;
            size_t sidx = ((size_t)f * LDIM + l) * TL + drow;
            float skr = sky_re[sidx];
            float ski = sky_im[sidx];
            if (isIm) {
                atomicAdd(&Sim_[f * SPITCH + drow], be_re * ski - be_im * skr);
            } else {
                atomicAdd(&Sre[f * SPITCH + drow], be_re * skr + be_im * ski);
                if (l == 0 && drow == LMAX) nrm[f] = be_re;   // beam monopole
            }
        }
    }
}

// ---------------- kernel 2: batched WMMA GEMM + phase/conj/sky reduction ------
// One workgroup per l.  Y[m,col] = sum_n dl[l,m,n] * X[n,col], X = beam*pg,
// via V_WMMA_F32_16X16X4_F32.  X is stored TRANSPOSED in LDS (Xt[col][n]) so
// each WMMA B operand is one contiguous, bank-conflict-free ds_load_b64.
__global__ __launch_bounds__(256) void sim_beamconv_kernel(
    const float* __restrict__ dl,
    const float* __restrict__ beam_re, const float* __restrict__ beam_im,
    const float* __restrict__ sky_re,  const float* __restrict__ sky_im,
    const float* __restrict__ euler,
    float* __restrict__ Sre, float* __restrict__ Sim_, float* __restrict__ nrm)
{
    __shared__ float Xt[32 * XTP];          // 32 cols x (K padded), transposed

    const int l      = blockIdx.x;
    const int tid    = threadIdx.x;
    const int lane   = tid & 31;
    const int wave   = tid >> 5;
    const int laneLo = lane & 15;
    const int laneHi = lane >> 4;

    const float alpha = euler[0];
    const float gamma = euler[2];

    for (int chunk = 0; chunk < 4; ++chunk) {
        // ---- stage Xt[c][n] for this 32-column chunk into LDS (zero-padded) ----
        for (int idx = tid; idx < 364 * 32; idx += 256) {
            int n    = idx >> 5;
            int c    = idx & 31;
            int gcol = chunk * 32 + c;
            float x = 0.0f;
            if (n < TL && gcol < NCOL) {
                int f = gcol >> 1;
                size_t bidx = ((size_t)f * LDIM + l) * TL + n;
                float br = beam_re[bidx];
                float bi = beam_im[bidx];
                float sn, cs;
                __sincosf((float)(n - LMAX) * gamma, &sn, &cs);
                // pg = cos - i sin:  Xre = br*cs + bi*sn ; Xim = bi*cs - br*sn
                x = (gcol & 1) ? (bi * cs - br * sn) : (br * cs + bi * sn);
            }
            Xt[c * XTP + n] = x;
        }
        __syncthreads();

        for (int mt = wave; mt < MT; mt += 8) {
            int m0   = mt * 16;
            int arow = min(m0 + laneLo, TL - 1);        // clamp: garbage rows discarded
            const float* dlrow = dl + ((size_t)l * TL + arow) * TL;
            __builtin_prefetch(dlrow, 0, 3);            // global_prefetch_b8

            const int kbase = 2 * laneHi;               // even -> 8B-aligned LDS b64
            const float* xrowA = Xt + (size_t)laneLo        * XTP;  // chunk cols 0..15
            const float* xrowB = Xt + (size_t)(16 + laneLo) * XTP;  // chunk cols 16..31

            v8f acc0 = {0.f,0.f,0.f,0.f,0.f,0.f,0.f,0.f};
            v8f acc1 = {0.f,0.f,0.f,0.f,0.f,0.f,0.f,0.f};

            // main K loop: kk=0..356, max k touched = 359 < 361 -> no checks
            for (int kk = 0; kk <= 356; kk += 4) {
                int k0 = kk + kbase;
                v2f a, b0, b1;
                a.x = dlrow[k0];
                a.y = dlrow[k0 + 1];
                b0  = *(const v2f*)(xrowA + k0);        // one ds_load_b64
                b1  = *(const v2f*)(xrowB + k0);        // one ds_load_b64
                acc0 = __builtin_amdgcn_wmma_f32_16x16x4_f32(
                           false, a, false, b0, (short)0, acc0, false, false);
                acc1 = __builtin_amdgcn_wmma_f32_16x16x4_f32(
                           false, a, false, b1, (short)0, acc1, false, false);
            }
            // K tail (kk=360): k in {360..363}; B rows >=361 are zero in LDS,
            // so clamped (in-bounds) A values only multiply zeros.
            {
                int k0  = 360 + kbase;
                int kc0 = min(k0,     TL - 1);
                int kc1 = min(k0 + 1, TL - 1);
                v2f a, b0, b1;
                a.x = dlrow[kc0];
                a.y = dlrow[kc1];
                b0  = *(const v2f*)(xrowA + k0);
                b1  = *(const v2f*)(xrowB + k0);
                acc0 = __builtin_amdgcn_wmma_f32_16x16x4_f32(
                           false, a, false, b0, (short)0, acc0, false, false);
                acc1 = __builtin_amdgcn_wmma_f32_16x16x4_f32(
                           false, a, false, b1, (short)0, acc1, false, false);
            }

            sim_epilogue(acc0, m0, laneHi, chunk * 32 + laneLo, l, alpha,
                         sky_re, sky_im, Sre, Sim_, nrm);
            sim_epilogue(acc1, m0, laneHi, chunk * 32 + 16 + laneLo, l, alpha,
                         sky_re, sky_im, Sre, Sim_, nrm);
        }
        __syncthreads();
    }
}

// ---------------- kernel 3: time-phase dot product + normalization ------------
__global__ __launch_bounds__(256) void sim_vis_kernel(
    const float* __restrict__ Sre, const float* __restrict__ Sim_,
    const float* __restrict__ nrm, const float* __restrict__ times,
    const float* __restrict__ fsky, const float* __restrict__ Tgnd,
    float* __restrict__ out)
{
    int idx = blockIdx.x * blockDim.x + threadIdx.x;
    if (idx >= NT * NF) return;
    int t = idx / NF;
    int f = idx % NF;
    float phi = 6.3003877373f * times[t];    // 2*pi*1.00273790935 (sidereal)
    float acc = 0.0f;
    for (int mi = 0; mi < TL; ++mi) {
        float sn, cs;
        __sincosf((float)(mi - LMAX) * phi, &sn, &cs);
        acc += cs * Sre[f * SPITCH + mi] - sn * Sim_[f * SPITCH + mi];
    }
    float norm = 3.5449077018f * nrm[f];     // sqrt(4*pi) * Re(beam_eq[f,0,lmax])
    float vs   = acc / norm;
    float fs   = fsky[f];
    out[idx]   = fs * vs + (1.0f - fs) * 300.0f * Tgnd[0];
}

// ---------------- host-side launch -------------------------------------------
extern "C" void kernel_launch(void* const* d_in, const int* in_sizes, int n_in,
                              void* d_out, int out_size, void* d_ws, size_t ws_size,
                              hipStream_t stream) {
    const float* beam_re = (const float*)d_in[0];
    const float* beam_im = (const float*)d_in[1];
    const float* sky_re  = (const float*)d_in[2];
    const float* sky_im  = (const float*)d_in[3];
    const float* dl      = (const float*)d_in[4];
    const float* euler   = (const float*)d_in[5];
    const float* times   = (const float*)d_in[6];
    const float* fsky    = (const float*)d_in[7];
    const float* Tgnd    = (const float*)d_in[8];
    (void)in_sizes; (void)n_in; (void)out_size; (void)ws_size;

    float* ws  = (float*)d_ws;
    float* Sre = ws;
    float* Sim = ws + (size_t)NF * SPITCH;
    float* nrm = ws + (size_t)2 * NF * SPITCH;

    int nzero = 2 * NF * SPITCH + NF;
    sim_zero_kernel<<<(nzero + 255) / 256, 256, 0, stream>>>(ws, nzero);
    sim_beamconv_kernel<<<LDIM, 256, 0, stream>>>(
        dl, beam_re, beam_im, sky_re, sky_im, euler, Sre, Sim, nrm);
    sim_vis_kernel<<<(NT * NF + 255) / 256, 256, 0, stream>>>(
        Sre, Sim, nrm, times, fsky, Tgnd, (float*)d_out);
}